// WKV_61581241090300
// MI455X (gfx1250) — compile-verified
//
#include <hip/hip_runtime.h>
#include <math.h>

// Problem constants (reference: B=4, T=2048, H=2048, fp32)
#define B_ 4
#define T_ 2048
#define H_ 2048
#define CHUNKS 32
#define LCH (T_ / CHUNKS)   /* 64 timesteps per chunk */
#define TILE 8              /* timesteps staged per tile */
#define NTILE (LCH / TILE)  /* 8 tiles per chunk */
#define BLK 256
#define NW (BLK / 32)       /* 8 waves per block (wave32) */
#define BH_ (B_ * H_)
#define CBH_ (CHUNKS * B_ * H_)

#if defined(__HIP_DEVICE_COMPILE__) &&                      \
    __has_builtin(__builtin_amdgcn_tensor_load_to_lds) &&   \
    __has_builtin(__builtin_amdgcn_s_wait_tensorcnt)
#define USE_TDM 1
#else
#define USE_TDM 0
#endif

#if !USE_TDM && defined(__HIP_DEVICE_COMPILE__) && \
    __has_builtin(__builtin_amdgcn_global_load_async_to_lds_b32)
#define USE_ASYNC 1
#else
#define USE_ASYNC 0
#endif

#define USE_LDS (USE_TDM || USE_ASYNC)

#if USE_LDS
typedef __attribute__((address_space(1))) int GINT;  // global int
typedef __attribute__((address_space(3))) int LINT;  // LDS int
#endif

#if USE_TDM
typedef unsigned int u32x4 __attribute__((ext_vector_type(4)));
typedef int i32x8 __attribute__((ext_vector_type(8)));
typedef int i32x4 __attribute__((ext_vector_type(4)));

#define TD1 (1u << 24) /* generous tensor_dim1 (rows) to avoid OOB clamping */

// Issue one TDM load: 2D tile, TILE rows x 32 floats, global row stride H_,
// contiguous into LDS at lds_off. Descriptor per CDNA5 ISA D# groups 0/1.
__device__ __forceinline__ void tdm_tile(const float* gtile, unsigned lds_off) {
  unsigned long long ga = (unsigned long long)gtile;
  unsigned lo = (unsigned)__builtin_amdgcn_readfirstlane((int)(unsigned)ga);
  unsigned hi = (unsigned)__builtin_amdgcn_readfirstlane((int)(unsigned)(ga >> 32));
  unsigned lds = (unsigned)__builtin_amdgcn_readfirstlane((int)lds_off);
  u32x4 g0;
  g0.x = 1u;                                   // count=1, user mode, no gather
  g0.y = lds;                                  // lds_addr (bytes)
  g0.z = lo;                                   // global_addr[31:0]
  g0.w = (hi & 0x01FFFFFFu) | 0x80000000u;     // global_addr[56:32] | type=2
  i32x8 g1;
  g1[0] = 0x20000;                             // data_size=2 (4 bytes)
  g1[1] = (int)(((unsigned)H_ & 0xFFFFu) << 16);            // tensor_dim0 lo16
  g1[2] = (int)(((unsigned)H_ >> 16) | ((TD1 & 0xFFFFu) << 16)); // dim0 hi|dim1 lo
  g1[3] = (int)((TD1 >> 16) | (32u << 16));    // dim1 hi16 | tile_dim0=32
  g1[4] = TILE;                                // tile_dim1=8, tile_dim2=0
  g1[5] = H_;                                  // tensor_dim0_stride lo32
  g1[6] = 0;                                   // stride hi16 | dim1_stride lo16
  g1[7] = 0;                                   // dim1_stride hi32
  i32x4 gz4 = {0, 0, 0, 0};
  i32x8 gz8 = {0, 0, 0, 0, 0, 0, 0, 0};
  __builtin_amdgcn_tensor_load_to_lds(g0, g1, gz4, gz4, gz8, 0);
}

#define STAGE_TILE(gk, gv, lk, lv, lane) \
  do {                                   \
    tdm_tile((gk), (lk));                \
    tdm_tile((gv), (lv));                \
  } while (0)
#define WAIT_STAGE(n)                          \
  do {                                         \
    __builtin_amdgcn_s_wait_tensorcnt(n);      \
    asm volatile("" ::: "memory");             \
  } while (0)
#define LDSOFF(p) ((unsigned)(unsigned long long)(LINT*)(p))

#elif USE_ASYNC

#if __has_builtin(__builtin_amdgcn_s_wait_asynccnt)
#define WAIT_STAGE(n)                          \
  do {                                         \
    __builtin_amdgcn_s_wait_asynccnt(n);       \
    asm volatile("" ::: "memory");             \
  } while (0)
#else
#define WAIT_STAGE(n) asm volatile("s_wait_asynccnt %0" ::"i"(n) : "memory")
#endif

// Per-lane async staging of a TILE x 32 tile (lane = column).
__device__ __forceinline__ void async_tile(const float* gtile, float* ltile,
                                           int lane) {
#pragma unroll
  for (int r = 0; r < TILE; ++r) {
    __builtin_amdgcn_global_load_async_to_lds_b32(
        (GINT*)(gtile + (long long)r * H_ + lane),
        (LINT*)(ltile + r * 32 + lane), 0, 0);
  }
}
#define STAGE_TILE(gk, gv, lk, lv, lane) \
  do {                                   \
    async_tile((gk), (lk), (lane));      \
    async_tile((gv), (lv), (lane));      \
  } while (0)
#define LDSOFF(p) (p)
#endif

__device__ __forceinline__ void wkv_step(float kt, float vt, float decay,
                                         float& num, float& den, float& norm) {
  float dpn = decay + norm;
  float nn = fmaxf(dpn, kt);
  float ed = __expf(dpn - nn);
  float ek = __expf(kt - nn);
  num = ed * num + ek * vt;
  den = ed * den + ek;
  norm = nn;
}

// ---------------- Pass 1: per-chunk local state (num, den, norm) ------------
// g = c*B*H + b*H + h ; lanes consecutive in h (coalesced 128B rows).
__global__ __launch_bounds__(BLK) void wkv_pass1(
    const float* __restrict__ k, const float* __restrict__ v,
    const float* __restrict__ time_decay, float* __restrict__ summ) {
#if USE_LDS
  __shared__ float sk[2][NW][TILE * 32];
  __shared__ float sv[2][NW][TILE * 32];
#endif
  const int tid = threadIdx.x;
  const int g = blockIdx.x * BLK + tid;
  const int h = g & (H_ - 1);
  const float decay = -__expf(time_decay[h]);

  float num = 0.f, den = 0.f, norm = -__builtin_huge_valf();

#if USE_LDS
  const int wid = __builtin_amdgcn_readfirstlane((int)(tid >> 5));
  const int lane = tid & 31;
  // Uniform (per-wave) tile base: b,c from block, column base = h of lane 0.
  const int bcU = (blockIdx.x * BLK) >> 11;
  const int bU = bcU & (B_ - 1);
  const int cU = bcU >> 2;
  const long long wbase =
      ((long long)bU * T_ + (long long)cU * LCH) * H_ +
      ((blockIdx.x * BLK) & (H_ - 1)) + (wid << 5);

  STAGE_TILE(k + wbase, v + wbase, LDSOFF(&sk[0][wid][0]),
             LDSOFF(&sv[0][wid][0]), lane);
#pragma unroll 1
  for (int t = 0; t < NTILE; ++t) {
    if (t + 1 < NTILE) {
      const long long ofs = wbase + (long long)(t + 1) * TILE * H_;
      STAGE_TILE(k + ofs, v + ofs, LDSOFF(&sk[(t + 1) & 1][wid][0]),
                 LDSOFF(&sv[(t + 1) & 1][wid][0]), lane);
      WAIT_STAGE(2 * (USE_TDM ? 1 : TILE));  // current tile complete (in-order)
    } else {
      WAIT_STAGE(0);
    }
    const float* bk = &sk[t & 1][wid][0];
    const float* bv = &sv[t & 1][wid][0];
#pragma unroll
    for (int r = 0; r < TILE; ++r)
      wkv_step(bk[r * 32 + lane], bv[r * 32 + lane], decay, num, den, norm);
  }
#else
  const int bc = g >> 11;
  const int b = bc & (B_ - 1);
  const int c = bc >> 2;
  const long long base = ((long long)b * T_ + (long long)c * LCH) * H_ + h;
#pragma unroll 4
  for (int j = 0; j < LCH; ++j)
    wkv_step(k[base + (long long)j * H_], v[base + (long long)j * H_], decay,
             num, den, norm);
#endif
  summ[g] = num;
  summ[g + CBH_] = den;
  summ[g + 2 * CBH_] = norm;
}

// -------- Pass 2: exclusive combine of chunk summaries per (b,h) ------------
__global__ __launch_bounds__(BLK) void wkv_pass2(
    const float* __restrict__ time_decay, const float* __restrict__ summ,
    float* __restrict__ pref, float* __restrict__ outTail) {
  const int g = blockIdx.x * BLK + threadIdx.x;  // [0, B*H)
  const int h = g & (H_ - 1);
  const float decay = -__expf(time_decay[h]);
  const float Ld = (float)LCH * decay;  // log-scale decay across a full chunk

  float num = 0.f, den = 0.f, norm = -__builtin_huge_valf();
#pragma unroll 1
  for (int c = 0; c < CHUNKS; ++c) {
    const int idx = c * BH_ + g;
    pref[idx] = num;  // state entering chunk c
    pref[idx + CBH_] = den;
    pref[idx + 2 * CBH_] = norm;
    float n1 = summ[idx];
    float d1 = summ[idx + CBH_];
    float m1 = summ[idx + 2 * CBH_];
    float a = Ld + norm;  // decayed prior log-magnitude
    float m = fmaxf(a, m1);
    float e0 = __expf(a - m);
    float e1 = __expf(m1 - m);
    num = e0 * num + e1 * n1;
    den = e0 * den + e1 * d1;
    norm = m;
  }
  outTail[g] = num;  // final carries: num, den, norm
  outTail[g + BH_] = den;
  outTail[g + 2 * BH_] = norm;
}

__device__ __forceinline__ void wkv_out_step(float kt, float vt, float wv,
                                             float decay, float& num,
                                             float& den, float& norm,
                                             float* optr) {
  // output from state *before* update
  float ct = wv + kt;
  float m = fmaxf(ct, norm);
  float ep = __expf(norm - m);
  float ec = __expf(ct - m);
  float o = (ep * num + ec * vt) / (ep * den + ec);
  __builtin_nontemporal_store(o, optr);
  // state update
  wkv_step(kt, vt, decay, num, den, norm);
}

// -------- Pass 3: rescan chunks from prefix states, write outputs ----------
__global__ __launch_bounds__(BLK) void wkv_pass3(
    const float* __restrict__ k, const float* __restrict__ v,
    const float* __restrict__ time_decay, const float* __restrict__ wtok,
    const float* __restrict__ pref, float* __restrict__ out) {
#if USE_LDS
  __shared__ float sk[2][NW][TILE * 32];
  __shared__ float sv[2][NW][TILE * 32];
#endif
  const int tid = threadIdx.x;
  const int g = blockIdx.x * BLK + tid;
  const int h = g & (H_ - 1);
  const float decay = -__expf(time_decay[h]);
  const float wv = wtok[h];

  float num = pref[g];
  float den = pref[g + CBH_];
  float norm = pref[g + 2 * CBH_];

#if USE_LDS
  const int wid = __builtin_amdgcn_readfirstlane((int)(tid >> 5));
  const int lane = tid & 31;
  const int bcU = (blockIdx.x * BLK) >> 11;
  const int bU = bcU & (B_ - 1);
  const int cU = bcU >> 2;
  const long long wbase =
      ((long long)bU * T_ + (long long)cU * LCH) * H_ +
      ((blockIdx.x * BLK) & (H_ - 1)) + (wid << 5);
  const long long tbase = wbase + lane;  // this thread's element base

  STAGE_TILE(k + wbase, v + wbase, LDSOFF(&sk[0][wid][0]),
             LDSOFF(&sv[0][wid][0]), lane);
#pragma unroll 1
  for (int t = 0; t < NTILE; ++t) {
    if (t + 1 < NTILE) {
      const long long ofs = wbase + (long long)(t + 1) * TILE * H_;
      STAGE_TILE(k + ofs, v + ofs, LDSOFF(&sk[(t + 1) & 1][wid][0]),
                 LDSOFF(&sv[(t + 1) & 1][wid][0]), lane);
      WAIT_STAGE(2 * (USE_TDM ? 1 : TILE));
    } else {
      WAIT_STAGE(0);
    }
    const float* bk = &sk[t & 1][wid][0];
    const float* bv = &sv[t & 1][wid][0];
#pragma unroll
    for (int r = 0; r < TILE; ++r)
      wkv_out_step(bk[r * 32 + lane], bv[r * 32 + lane], wv, decay, num, den,
                   norm, out + tbase + (long long)(t * TILE + r) * H_);
  }
#else
  const int bc = g >> 11;
  const int b = bc & (B_ - 1);
  const int c = bc >> 2;
  const long long base = ((long long)b * T_ + (long long)c * LCH) * H_ + h;
#pragma unroll 4
  for (int j = 0; j < LCH; ++j)
    wkv_out_step(k[base + (long long)j * H_], v[base + (long long)j * H_], wv,
                 decay, num, den, norm, out + base + (long long)j * H_);
#endif
}

// -------- Fallback: monolithic per-(b,h) scan (used if ws too small) --------
__global__ __launch_bounds__(BLK) void wkv_mono(
    const float* __restrict__ k, const float* __restrict__ v,
    const float* __restrict__ time_decay, const float* __restrict__ wtok,
    float* __restrict__ out) {
  const int g = blockIdx.x * BLK + threadIdx.x;  // [0, B*H)
  const int h = g & (H_ - 1);
  const int b = g >> 11;
  const float decay = -__expf(time_decay[h]);
  const float wv = wtok[h];
  const long long base = (long long)b * T_ * H_ + h;
  float num = 0.f, den = 0.f, norm = -__builtin_huge_valf();
#pragma unroll 1
  for (int t = 0; t < T_; ++t)
    wkv_out_step(k[base + (long long)t * H_], v[base + (long long)t * H_], wv,
                 decay, num, den, norm, out + base + (long long)t * H_);
  const size_t BTH = (size_t)B_ * T_ * H_;
  out[BTH + g] = num;
  out[BTH + BH_ + g] = den;
  out[BTH + 2 * BH_ + g] = norm;
}

extern "C" void kernel_launch(void* const* d_in, const int* in_sizes, int n_in,
                              void* d_out, int out_size, void* d_ws,
                              size_t ws_size, hipStream_t stream) {
  const float* k = (const float*)d_in[0];
  const float* v = (const float*)d_in[1];
  const float* td = (const float*)d_in[2];
  const float* wt = (const float*)d_in[3];
  float* out = (float*)d_out;

  const size_t need = (size_t)6 * CBH_ * sizeof(float);  // summaries + prefixes
  if (ws_size >= need) {
    float* summ = (float*)d_ws;             // 3 planes of CBH_
    float* pref = summ + 3 * (size_t)CBH_;  // 3 planes of CBH_
    wkv_pass1<<<(B_ * H_ * CHUNKS) / BLK, BLK, 0, stream>>>(k, v, td, summ);
    wkv_pass2<<<(B_ * H_) / BLK, BLK, 0, stream>>>(
        td, summ, pref, out + (size_t)B_ * T_ * H_);
    wkv_pass3<<<(B_ * H_ * CHUNKS) / BLK, BLK, 0, stream>>>(k, v, td, wt, pref,
                                                            out);
  } else {
    wkv_mono<<<(B_ * H_) / BLK, BLK, 0, stream>>>(k, v, td, wt, out);
  }
}